// Encoder_8452495638861
// MI455X (gfx1250) — compile-verified
//
#include <hip/hip_runtime.h>
#include <hip/hip_bf16.h>
#include <stddef.h>

// ---------------------------------------------------------------------------
// GIN encoder for MI455X (gfx1250, wave32).
//   h0 = x @ W_in^T + b_in
//   for l in 0..2:  agg = segment_sum(h_l[src], dst);  h_{l+1} = (h_l+agg) @ W_l^T + b_l
//   out = concat(h0..h3) @ W_out^T + b_out
// Dense GEMMs: V_WMMA_F32_16X16X4_F32 (fp32, matches reference precision class).
// ---------------------------------------------------------------------------

#define N_NODES 50000
#define N_EDGES 800000
#define IN_DIM  128
#define HID     96
#define OUT_DIM 128
#define N_LAYERS 3
#define CAT_DIM ((N_LAYERS + 1) * HID)   // 384
#define MT 5                              // M tiles per wave (3125 = 5 * 625)

typedef float v2f __attribute__((ext_vector_type(2)));
typedef float v8f __attribute__((ext_vector_type(8)));

// ---------------------------------------------------------------------------
// WMMA f32 GEMM:  D[m, n] = sum_k (A[m,k] (+A2[m,k])) * W[n,k] + bias[n]
// All shapes compile-time; 32-bit lane offsets off uniform bases (saddr+voffset).
// Block: (N/16) waves; wave w owns n-tile n0=16w and MT m-tiles.
// EXEC all-ones by construction (dims divide exactly).
// ---------------------------------------------------------------------------
template <int K, int LDA, int LDA2, int LDD, bool HAS_ADD>
__global__ void wmma_gemm_f32(const float* __restrict__ A,
                              const float* __restrict__ A2,
                              const float* __restrict__ W,
                              const float* __restrict__ bias,
                              float* __restrict__ D) {
  const int wave = threadIdx.x >> 5;
  const int n0   = wave * 16;
  const int lane = threadIdx.x & 31;
  const int half = lane >> 4;          // 0: K sub-cols {0,1}; 1: {2,3}
  const int l16  = lane & 15;
  const int m0   = blockIdx.x * (16 * MT);

  // Per-lane 32-bit offsets (max < 2^25 elements -> int is safe).
  int aoff[MT];
  int a2off[MT];
#pragma unroll
  for (int t = 0; t < MT; ++t) {
    aoff[t] = (m0 + t * 16 + l16) * LDA + 2 * half;
    if (HAS_ADD) a2off[t] = (m0 + t * 16 + l16) * LDA2 + 2 * half;
  }
  const int woff = (n0 + l16) * K + 2 * half;

  v8f acc[MT] = {};

#pragma unroll 2
  for (int k = 0; k < K; k += 4) {
    v2f b;
    b.x = W[woff + k];
    b.y = W[woff + k + 1];
#pragma unroll
    for (int t = 0; t < MT; ++t) {
      v2f a;
      a.x = A[aoff[t] + k];
      a.y = A[aoff[t] + k + 1];
      if (HAS_ADD) {
        a.x += A2[a2off[t] + k];
        a.y += A2[a2off[t] + k + 1];
      }
      // D = A(16x4) * B(4x16) + C  -> v_wmma_f32_16x16x4_f32
      acc[t] = __builtin_amdgcn_wmma_f32_16x16x4_f32(
          /*neg_a=*/false, a, /*neg_b=*/false, b,
          /*c_mod=*/(short)0, acc[t], /*reuse_a=*/false, /*reuse_b=*/false);
    }
  }

  // C/D layout: VGPR r holds M = r + 8*half, N = l16.
  const float bv = bias[n0 + l16];
#pragma unroll
  for (int t = 0; t < MT; ++t) {
#pragma unroll
    for (int r = 0; r < 8; ++r) {
      D[(m0 + t * 16 + r + 8 * half) * LDD + n0 + l16] = acc[t][r] + bv;
    }
  }
}

// ---------------------------------------------------------------------------
// Zero a float buffer (float4 granularity; count % 4 == 0).
// ---------------------------------------------------------------------------
__global__ void zero_f32(float* __restrict__ p, int count4) {
  int i = blockIdx.x * blockDim.x + threadIdx.x;
  if (i < count4) {
    ((float4*)p)[i] = make_float4(0.f, 0.f, 0.f, 0.f);
  }
}

// ---------------------------------------------------------------------------
// Edge scatter-add: agg[dst[e], :] += h[src[e], :]  (h row stride CAT_DIM)
// One thread per (edge, 4-feature chunk): float4 gather, 4 f32 atomics.
// ---------------------------------------------------------------------------
__global__ void scatter_add_edges(const float* __restrict__ h,
                                  const int* __restrict__ src,
                                  const int* __restrict__ dst,
                                  float* __restrict__ agg) {
  const int tid = blockIdx.x * blockDim.x + threadIdx.x;
  const int e = tid / (HID / 4);
  if (e >= N_EDGES) return;
  const int c = (tid % (HID / 4)) * 4;

  const int s = src[e];
  const int d = dst[e];
  const float4 v = *(const float4*)(h + s * CAT_DIM + c);
  float* out = agg + d * HID + c;
  atomicAdd(out + 0, v.x);
  atomicAdd(out + 1, v.y);
  atomicAdd(out + 2, v.z);
  atomicAdd(out + 3, v.w);
}

// ---------------------------------------------------------------------------
// Launch
// ---------------------------------------------------------------------------
extern "C" void kernel_launch(void* const* d_in, const int* in_sizes, int n_in,
                              void* d_out, int out_size, void* d_ws, size_t ws_size,
                              hipStream_t stream) {
  (void)in_sizes; (void)n_in; (void)out_size; (void)ws_size;

  const float* x        = (const float*)d_in[0];                 // [N_NODES, IN_DIM]
  const int*   edge_idx = (const int*)d_in[1];                   // [2, N_EDGES]
  const float* W_in     = (const float*)d_in[2];                 // [HID, IN_DIM]
  const float* b_in     = (const float*)d_in[3];                 // [HID]
  const float* W_layers = (const float*)d_in[4];                 // [L, HID, HID]
  const float* b_layers = (const float*)d_in[5];                 // [L, HID]
  const float* W_out    = (const float*)d_in[6];                 // [OUT_DIM, CAT_DIM]
  const float* b_out    = (const float*)d_in[7];                 // [OUT_DIM]
  float*       out      = (float*)d_out;                         // [N_NODES, OUT_DIM]

  const int* src = edge_idx;
  const int* dst = edge_idx + N_EDGES;

  // Workspace: cat [N_NODES x CAT_DIM] f32, then agg [N_NODES x HID] f32 (~96 MB).
  float* cat = (float*)d_ws;
  float* agg = cat + (size_t)N_NODES * CAT_DIM;

  const int mBlocks = N_NODES / 16 / MT;  // 625, exact

  // 1) h0 = x @ W_in^T + b_in  -> cat[:, 0:96]
  wmma_gemm_f32<IN_DIM, IN_DIM, 0, CAT_DIM, false>
      <<<mBlocks, (HID / 16) * 32, 0, stream>>>(x, nullptr, W_in, b_in, cat);

  // 2) GIN layers
  const int zeroBlocks    = (N_NODES * HID / 4 + 255) / 256;
  const int scatterBlocks = (N_EDGES * (HID / 4) + 255) / 256;
  for (int l = 0; l < N_LAYERS; ++l) {
    const float* h_l = cat + (size_t)l * HID;  // row stride CAT_DIM
    zero_f32<<<zeroBlocks, 256, 0, stream>>>(agg, N_NODES * HID / 4);
    scatter_add_edges<<<scatterBlocks, 256, 0, stream>>>(h_l, src, dst, agg);
    wmma_gemm_f32<HID, CAT_DIM, HID, CAT_DIM, true>
        <<<mBlocks, (HID / 16) * 32, 0, stream>>>(
            h_l, agg, W_layers + (size_t)l * HID * HID,
            b_layers + (size_t)l * HID, cat + (size_t)(l + 1) * HID);
  }

  // 3) out = cat @ W_out^T + b_out
  wmma_gemm_f32<CAT_DIM, CAT_DIM, 0, OUT_DIM, false>
      <<<mBlocks, (OUT_DIM / 16) * 32, 0, stream>>>(cat, nullptr, W_out, b_out, out);
}